// K_Loss_56375740727688
// MI455X (gfx1250) — compile-verified
//
#include <hip/hip_runtime.h>
#include <hip/hip_bf16.h>
#include <math.h>

// ---------------------------------------------------------------------------
// result = mean_b( trace_b / 128 ) with
//   trace_b = tr(G) - 2k * sum_n [ cos(t0(n))*G(n,n+x) + cos(t1(n))*G(n,n+y) ]
//   G = L^T L,  L = 128x128 lower-triangular (packed tril in net_out[b])
//   n+x = (k+16) mod 128 ; n+y = k with (k%16 + 2) mod 16
// Only 16 of the 64 16x16 tiles of G are needed: 8 diagonal tiles and the
// 8 block-superdiagonal tiles (a, a+1 mod 8). Each tile is computed with
// fp32 WMMA: 32 x v_wmma_f32_16x16x4_f32 chained over K = 128.
// ---------------------------------------------------------------------------

typedef float v2f __attribute__((ext_vector_type(2)));
typedef float v8f __attribute__((ext_vector_type(8)));

#define NMAT   128
#define NTRI   8256          // 128*129/2
#define LTS    130           // padded column stride (words, even -> 8B aligned)
#define KAPPA2 0.552f        // 2 * 0.276

__global__ __launch_bounds__(512)
void wdirac_trace_kernel(const float* __restrict__ net_out,  // (B, 8256)
                         const float* __restrict__ U1,       // (B, 2, 8, 8)
                         float* __restrict__ traces)         // (B,)
{
    extern __shared__ __align__(16) float smem[];
    float* LT  = smem;                  // L^T column-major: LT[k*LTS + i] = L[i][k]
    float* cs  = smem + NMAT * LTS;     // 128 cosines: [mu*64 + nx*8 + ny]
    float* red = cs + 128;              // 512-float reduction scratch

    const int b   = blockIdx.x;
    const int tid = threadIdx.x;

    // ---- stage L^T (zeros above diagonal) and the link cosines into LDS ----
    for (int idx = tid; idx < NMAT * LTS; idx += 512) LT[idx] = 0.0f;
    if (tid < 128) cs[tid] = cosf(U1[(size_t)b * 128 + tid]);
    __syncthreads();

    const float* src = net_out + (size_t)b * NTRI;
    for (int p = tid; p < NTRI; p += 512) {
        // invert p = r*(r+1)/2 + c
        int r = (int)((sqrtf(8.0f * (float)p + 1.0f) - 1.0f) * 0.5f);
        while ((r + 1) * (r + 2) / 2 <= p) ++r;
        while (r * (r + 1) / 2 > p)       --r;
        const int c = p - r * (r + 1) / 2;
        LT[c * LTS + r] = src[p];       // L[r][c] -> LT[c][r]
    }
    __syncthreads();

    // ---- per-wave 16x16 Gram tile via fp32 WMMA --------------------------
    const int lane = tid & 31;          // wave32
    const int w    = tid >> 5;          // wave id 0..15
    const int n    = lane & 15;         // fragment row/col index (A: M, B/C/D: N)
    const int hi   = lane >> 4;         // K-half (A/B) / M-half (C/D)
    const int a    = (w < 8) ? w : (w - 8);
    const int bc   = (w < 8) ? a : ((a + 1) & 7);   // column-block of the tile

    const float* colA = &LT[(16 * a  + n) * LTS];
    const float* colB = &LT[(16 * bc + n) * LTS];

    v8f acc = {0.f, 0.f, 0.f, 0.f, 0.f, 0.f, 0.f, 0.f};
#pragma unroll
    for (int step = 0; step < 32; ++step) {
        const int i0 = step * 4 + hi * 2;           // even -> 8B aligned
        const v2f av = *(const v2f*)(colA + i0);    // A[m][k0..k0+1], m = n
        const v2f bv = *(const v2f*)(colB + i0);    // B[k0..k0+1][n]
        acc = __builtin_amdgcn_wmma_f32_16x16x4_f32(
                  /*neg_a=*/false, av, /*neg_b=*/false, bv,
                  /*c_mod=*/(short)0, acc, /*reuse_a=*/false, /*reuse_b=*/false);
    }

    // ---- extract the needed entries from the C/D layout -------------------
    // lane holds D[m][n] in acc[v] with m = 8*hi + v, n = lane & 15.
    float part = 0.0f;
#pragma unroll
    for (int v = 0; v < 8; ++v) {
        const int   m  = hi * 8 + v;
        const float dv = acc[v];
        if (w < 8) {
            // diagonal tile (a,a): tr(G) part + y-hop entries D[m][(m+2)&15]
            if (m == n)              part += dv;
            if (n == ((m + 2) & 15)) part -= KAPPA2 * cs[64 + a * 8 + (m >> 1)] * dv;
        } else {
            // x-hop tile (a, a+1 mod 8): G(k, k+16) = D[m][m]
            if (m == n)              part -= KAPPA2 * cs[a * 8 + (m >> 1)] * dv;
        }
    }

    // ---- deterministic block reduction -> trace_b ------------------------
    red[tid] = part;
    __syncthreads();
    for (int s = 256; s > 0; s >>= 1) {
        if (tid < s) red[tid] += red[tid + s];
        __syncthreads();
    }
    if (tid == 0) traces[b] = red[0];
}

__global__ __launch_bounds__(256)
void reduce_traces_kernel(const float* __restrict__ traces,
                          float* __restrict__ out, int B)
{
    __shared__ float red[256];
    float s = 0.0f;
    for (int i = threadIdx.x; i < B; i += 256) s += traces[i];
    red[threadIdx.x] = s;
    __syncthreads();
    for (int st = 128; st > 0; st >>= 1) {
        if (threadIdx.x < st) red[threadIdx.x] += red[threadIdx.x + st];
        __syncthreads();
    }
    // trace2 is exactly N (=128) per batch: D(e_k) has unit diagonal.
    if (threadIdx.x == 0) out[0] = red[0] / ((float)B * 128.0f);
}

extern "C" void kernel_launch(void* const* d_in, const int* in_sizes, int n_in,
                              void* d_out, int out_size, void* d_ws, size_t ws_size,
                              hipStream_t stream)
{
    const float* net_out = (const float*)d_in[0];   // (B, 8256) f32
    const float* U1      = (const float*)d_in[1];   // (B, 2, 8, 8) f32
    float*       traces  = (float*)d_ws;            // B floats of scratch
    float*       out     = (float*)d_out;           // scalar

    const int B = in_sizes[0] / NTRI;

    const size_t shmem = (size_t)(NMAT * LTS + 128 + 512) * sizeof(float); // ~69 KB
    wdirac_trace_kernel<<<B, 512, shmem, stream>>>(net_out, U1, traces);
    reduce_traces_kernel<<<1, 256, 0, stream>>>(traces, out, B);
}